// GCNEncoder_66992899883196
// MI455X (gfx1250) — compile-verified
//
#include <hip/hip_runtime.h>

// ---------------------------------------------------------------------------
// 3-layer GCN for MI455X (gfx1250, wave32).
//   h   = x @ W                      -> v_wmma_f32_16x16x4_f32 (exact fp32)
//   agg = h*dinv^2 + b  (self loop + bias)
//   agg += dinv[s]*dinv[d]*h[s] over edges (fp32 L2 atomics; all buffers fit
//          in the 192MB L2, so the scatter is L2-resident, not HBM-bound)
//   act = relu(agg)  (layers 1,2); layer 3 accumulates into d_out directly.
// ---------------------------------------------------------------------------

typedef __attribute__((ext_vector_type(2))) float v2f;
typedef __attribute__((ext_vector_type(8))) float v8f;

#define FEAT 128

// ---- degree / normalization -----------------------------------------------
__global__ __launch_bounds__(256) void k_deg_init(float* __restrict__ deg, int n) {
    int i = blockIdx.x * blockDim.x + threadIdx.x;
    if (i < n) deg[i] = 1.0f;                       // self loop
}

__global__ __launch_bounds__(256) void k_deg_accum(const long long* __restrict__ dst,
                                                   float* __restrict__ deg, int E) {
    int e = blockIdx.x * blockDim.x + threadIdx.x;
    if (e < E) atomicAdd(&deg[dst[e]], 1.0f);
}

__global__ __launch_bounds__(256) void k_dinv(float* __restrict__ deg, int n) {
    int i = blockIdx.x * blockDim.x + threadIdx.x;
    if (i < n) deg[i] = rsqrtf(deg[i]);             // deg >= 1 always
}

// ---- fp32 WMMA GEMM: H[M x 128] = A[M x 128] * W[128 x 128] ----------------
// grid.x = M/16 (M must be a multiple of 16; N=50000 is), block = 256 = 8
// waves; wave w computes output tile rows [16*bx,16*bx+16) x cols [16*w,16*w+16).
__global__ __launch_bounds__(256) void k_gemm(const float* __restrict__ A,
                                              const float* __restrict__ W,
                                              float* __restrict__ H) {
    const int lane = threadIdx.x & 31;
    const int wave = threadIdx.x >> 5;
    const int l16  = lane & 15;
    const int hi   = lane >> 4;          // 0: K=0,1 half; 1: K=2,3 half
    const int m0   = blockIdx.x * 16;
    const int n0   = wave * 16;

    // A fragment base: lanes 0-15 hold (M=l16, K=k+0/k+1), lanes 16-31 hold K=k+2/k+3
    const float* arow = A + (size_t)(m0 + l16) * FEAT + hi * 2;
    // B fragment base: lanes 0-15 hold (K=k+0/k+1, N=n0+l16), lanes 16-31 K=k+2/k+3
    const float* bcol = W + (size_t)(hi * 2) * FEAT + n0 + l16;

    v8f acc = {};
#pragma unroll
    for (int k = 0; k < FEAT; k += 4) {
        v2f a = *(const v2f*)(arow + k);                 // contiguous pair -> b64 load
        v2f b;
        b.x = bcol[(size_t)k * FEAT];
        b.y = bcol[(size_t)(k + 1) * FEAT];
        acc = __builtin_amdgcn_wmma_f32_16x16x4_f32(
            /*neg_a=*/false, a, /*neg_b=*/false, b,
            /*c_mod=*/(short)0, acc, /*reuse_a=*/false, /*reuse_b=*/false);
    }

    // C/D layout: VGPR r -> (M = r + 8*hi, N = l16)
    float* orow = H + (size_t)(m0 + hi * 8) * FEAT + n0 + l16;
#pragma unroll
    for (int r = 0; r < 8; ++r) orow[(size_t)r * FEAT] = acc[r];
}

// ---- self-loop + bias init: out = h*dinv^2 + b -----------------------------
__global__ __launch_bounds__(256) void k_self_bias(const float* __restrict__ h,
                                                   const float* __restrict__ dinv,
                                                   const float* __restrict__ bias,
                                                   float* __restrict__ out, int n) {
    int i = blockIdx.x * blockDim.x + threadIdx.x;
    if (i < n * FEAT) {
        int node = i >> 7;
        float di = dinv[node];
        out[i] = h[i] * di * di + bias[i & (FEAT - 1)];
    }
}

// ---- edge scatter: agg[dst] += dinv[src]*dinv[dst]*h[src] ------------------
// 32 lanes per edge; lane covers features [4*lane, 4*lane+4) -> float4 gather
// of h[src] is fully coalesced; 4 fp32 atomics into agg[dst].
__global__ __launch_bounds__(256) void k_edge(const float* __restrict__ h,
                                              const long long* __restrict__ src,
                                              const long long* __restrict__ dst,
                                              const float* __restrict__ dinv,
                                              float* __restrict__ agg, int E) {
    long long i = (long long)blockIdx.x * blockDim.x + threadIdx.x;
    if (i >= (long long)E * 32) return;
    int e    = (int)(i >> 5);
    int lane = (int)(i & 31);
    long long s = src[e];
    long long d = dst[e];
    float nrm = dinv[s] * dinv[d];
    float4 hv = *(const float4*)(h + s * FEAT + lane * 4);
    float* op = agg + d * FEAT + lane * 4;
    atomicAdd(op + 0, hv.x * nrm);
    atomicAdd(op + 1, hv.y * nrm);
    atomicAdd(op + 2, hv.z * nrm);
    atomicAdd(op + 3, hv.w * nrm);
}

// ---- relu ------------------------------------------------------------------
__global__ __launch_bounds__(256) void k_relu(const float* __restrict__ in,
                                              float* __restrict__ out, int n) {
    int i = blockIdx.x * blockDim.x + threadIdx.x;
    if (i < n) out[i] = fmaxf(in[i], 0.0f);
}

// ---------------------------------------------------------------------------
extern "C" void kernel_launch(void* const* d_in, const int* in_sizes, int n_in,
                              void* d_out, int out_size, void* d_ws, size_t ws_size,
                              hipStream_t stream) {
    (void)n_in; (void)out_size; (void)ws_size;

    const float*     x   = (const float*)d_in[0];
    const long long* ei  = (const long long*)d_in[1];   // int64 edge_index [2,E]
    const float*     W1  = (const float*)d_in[2];
    const float*     b1  = (const float*)d_in[3];
    const float*     W2  = (const float*)d_in[4];
    const float*     b2  = (const float*)d_in[5];
    const float*     W3  = (const float*)d_in[6];
    const float*     b3  = (const float*)d_in[7];
    float*           out = (float*)d_out;

    const int N = in_sizes[0] / FEAT;   // 50000 (multiple of 16)
    const int E = in_sizes[1] / 2;      // 800000
    const long long* src = ei;
    const long long* dst = ei + E;

    // workspace carve-out
    char*  ws  = (char*)d_ws;
    size_t off = 0;
    auto carve = [&](size_t bytes) -> float* {
        float* p = (float*)(ws + off);
        off = (off + bytes + 255) & ~(size_t)255;
        return p;
    };
    float* dinv = carve((size_t)N * sizeof(float));
    float* h    = carve((size_t)N * FEAT * sizeof(float));
    float* agg  = carve((size_t)N * FEAT * sizeof(float));
    float* act  = carve((size_t)N * FEAT * sizeof(float));

    const dim3 blk(256);
    const int  nBlkN  = (N + 255) / 256;
    const int  nBlkE  = (E + 255) / 256;
    const int  nd     = N * FEAT;
    const int  nBlkND = (nd + 255) / 256;
    const int  nBlkEW = (int)(((long long)E * 32 + 255) / 256);
    const int  nBlkMM = N / 16;

    // normalization (shared by all layers)
    k_deg_init <<<nBlkN, blk, 0, stream>>>(dinv, N);
    k_deg_accum<<<nBlkE, blk, 0, stream>>>(dst, dinv, E);
    k_dinv     <<<nBlkN, blk, 0, stream>>>(dinv, N);

    // layer 1
    k_gemm     <<<nBlkMM, blk, 0, stream>>>(x, W1, h);
    k_self_bias<<<nBlkND, blk, 0, stream>>>(h, dinv, b1, agg, N);
    k_edge     <<<nBlkEW, blk, 0, stream>>>(h, src, dst, dinv, agg, E);
    k_relu     <<<nBlkND, blk, 0, stream>>>(agg, act, nd);

    // layer 2
    k_gemm     <<<nBlkMM, blk, 0, stream>>>(act, W2, h);
    k_self_bias<<<nBlkND, blk, 0, stream>>>(h, dinv, b2, agg, N);
    k_edge     <<<nBlkEW, blk, 0, stream>>>(h, src, dst, dinv, agg, E);
    k_relu     <<<nBlkND, blk, 0, stream>>>(agg, act, nd);

    // layer 3 (no relu, accumulate straight into d_out)
    k_gemm     <<<nBlkMM, blk, 0, stream>>>(act, W3, h);
    k_self_bias<<<nBlkND, blk, 0, stream>>>(h, dinv, b3, out, N);
    k_edge     <<<nBlkEW, blk, 0, stream>>>(h, src, dst, dinv, out, E);
}